// SpatialAttention_12489764897466
// MI455X (gfx1250) — compile-verified
//
#include <hip/hip_runtime.h>
#include <hip/hip_bf16.h>

typedef __attribute__((ext_vector_type(2))) float v2f;
typedef __attribute__((ext_vector_type(8))) float v8f;

#define S_LEN   2048
#define HID     512
#define NH      8
#define HD      64
#define TH3     1536   // 3*H
#define SPITCH  2056   // LDS row pitch (even -> 8B aligned v2f, 8-bank skew)

static __device__ __forceinline__ v8f wmma_f32(v2f a, v2f b, v8f c) {
  // V_WMMA_F32_16X16X4_F32 : D(16x16 f32) = A(16x4 f32) * B(4x16 f32) + C
  return __builtin_amdgcn_wmma_f32_16x16x4_f32(
      /*neg_a=*/false, a, /*neg_b=*/false, b,
      /*c_mod=*/(short)0, c, /*reuse_a=*/false, /*reuse_b=*/false);
}

// ---------------------------------------------------------------------------
// Register-blocked GEMM:  C[M,N] = A[M,K] * B[N,K]^T + bias[N]
// One wave per 32x32 output tile (2x2 grid of 16x16 WMMA accumulators):
// 4 WMMAs per 4 fragment loads, A/B frags reused across the 2x2 block.
// ---------------------------------------------------------------------------
__global__ __launch_bounds__(256) void gemm32_bias(
    const float* __restrict__ A, const float* __restrict__ Bm,
    const float* __restrict__ bias, float* __restrict__ C,
    int M, int N, int K)
{
  int wave = blockIdx.x * (blockDim.x >> 5) + (threadIdx.x >> 5);
  int ntn = N >> 5;
  int mt = wave / ntn;
  int nt = wave - mt * ntn;
  if (mt >= (M >> 5)) return;          // uniform per wave

  int lane = threadIdx.x & 31;
  int l    = lane & 15;
  int half = lane >> 4;                // 0: K 0..1, 1: K 2..3 of each step

  const float* a0 = A  + (size_t)(mt * 32 + l) * K + half * 2;
  const float* a1 = a0 + (size_t)16 * K;
  const float* b0 = Bm + (size_t)(nt * 32 + l) * K + half * 2;
  const float* b1 = b0 + (size_t)16 * K;

  v8f acc00 = {0.f,0.f,0.f,0.f,0.f,0.f,0.f,0.f};
  v8f acc01 = acc00, acc10 = acc00, acc11 = acc00;

#pragma unroll 8
  for (int k = 0; k < K; k += 4) {
    if ((k & 31) == 0) {               // one WGP-scope prefetch per 128B line
      __builtin_prefetch(a0 + k + 64, 0, 3);
      __builtin_prefetch(a1 + k + 64, 0, 3);
      __builtin_prefetch(b0 + k + 64, 0, 3);
      __builtin_prefetch(b1 + k + 64, 0, 3);
    }
    v2f va0 = *(const v2f*)(a0 + k);
    v2f va1 = *(const v2f*)(a1 + k);
    v2f vb0 = *(const v2f*)(b0 + k);
    v2f vb1 = *(const v2f*)(b1 + k);
    acc00 = wmma_f32(va0, vb0, acc00);
    acc01 = wmma_f32(va0, vb1, acc01);
    acc10 = wmma_f32(va1, vb0, acc10);
    acc11 = wmma_f32(va1, vb1, acc11);
  }

  float bv0 = bias[nt * 32 + l];
  float bv1 = bias[nt * 32 + 16 + l];
  float* c00 = C + (size_t)(mt * 32 + half * 8) * N + nt * 32 + l;
#pragma unroll
  for (int r = 0; r < 8; ++r) {
    c00[(size_t)r * N]            = acc00[r] + bv0;
    c00[(size_t)r * N + 16]       = acc01[r] + bv1;
    c00[(size_t)(r + 16) * N]     = acc10[r] + bv0;
    c00[(size_t)(r + 16) * N + 16] = acc11[r] + bv1;
  }
}

// ---------------------------------------------------------------------------
// Fused spatial attention for one (batch b, 16-query tile):
//   scores -> +bias -> mask -> softmax -> head-mean accum -> P@V
// Scores (16x2048) and head-mean (16x2048) stay resident in 320KB LDS.
// ---------------------------------------------------------------------------
__global__ __launch_bounds__(256) void attn_fused(
    const float* __restrict__ proj,      // [B,S,3H]: Q @ +0, K @ +512, V @ +1024
    const float* __restrict__ pos,       // [B,S,3]
    const int*   __restrict__ amask,     // [B,S]
    float* __restrict__ ctx,             // [B,S,H] attention output (pre out-proj)
    float* __restrict__ out2)            // [B,S,S] mean-over-heads attn weights
{
  extern __shared__ float smem[];
  float* sc = smem;                       // 16 x SPITCH  scores / weights
  float* mn = smem + 16 * SPITCH;         // 16 x SPITCH  mean accumulator
  float* red = mn + 16 * SPITCH;          // 4*256 cross-wave reduce scratch
  float* pq  = red + 1024;                // 16 x 3 query positions

  const int b    = blockIdx.y;
  const int q0   = blockIdx.x * 16;
  const int tid  = threadIdx.x;
  const int wave = tid >> 5;
  const int lane = tid & 31;
  const int l    = lane & 15;
  const int half = lane >> 4;

  for (int i = tid; i < 16 * SPITCH; i += 256) mn[i] = 0.f;
  if (tid < 48)
    pq[tid] = pos[(size_t)(b * S_LEN + q0 + tid / 3) * 3 + (tid % 3)];
  __syncthreads();

  for (int h = 0; h < NH; ++h) {
    // ---- Phase A: scores = (Q K^T)*0.125 - d2*0.125, masked --------------
    v2f qa[16];                            // Q fragments, K-dim 0..63
    const float* qrow = proj + (size_t)(b * S_LEN + q0 + l) * TH3 + h * HD + half * 2;
#pragma unroll
    for (int kk = 0; kk < 16; ++kk) qa[kk] = *(const v2f*)(qrow + kk * 4);

    for (int kt = wave; kt < S_LEN / 16; kt += 8) {
      int k0 = kt * 16;
      int ks = k0 + l;                     // this lane's key column
      const float* pk = pos + (size_t)(b * S_LEN + ks) * 3;
      float pk0 = pk[0], pk1 = pk[1], pk2 = pk[2];
      int mv = amask[b * S_LEN + ks];
      const float* kr = proj + (size_t)(b * S_LEN + ks) * TH3 + HID + h * HD + half * 2;

      v8f acc = {0.f,0.f,0.f,0.f,0.f,0.f,0.f,0.f};
#pragma unroll
      for (int kk = 0; kk < 16; ++kk) {
        v2f bf = *(const v2f*)(kr + kk * 4);
        acc = wmma_f32(qa[kk], bf, acc);
      }
#pragma unroll
      for (int r = 0; r < 8; ++r) {
        int m = r + half * 8;
        float dx = pq[m * 3 + 0] - pk0;
        float dy = pq[m * 3 + 1] - pk1;
        float dz = pq[m * 3 + 2] - pk2;
        // log(exp(-d2/(2*sigma^2))) == -d2/8 ; scale = hd^-0.5 = 0.125
        float s = acc[r] * 0.125f - (dx * dx + dy * dy + dz * dz) * 0.125f;
        s = mv ? s : -__builtin_inff();
        sc[m * SPITCH + ks] = s;
      }
    }
    __syncthreads();

    // ---- Phase B: row softmax + mean accumulation ------------------------
#pragma unroll
    for (int rr = wave * 2; rr < wave * 2 + 2; ++rr) {
      float* row = sc + rr * SPITCH;
      float mx = -__builtin_inff();
      for (int c = lane; c < S_LEN; c += 32) mx = fmaxf(mx, row[c]);
      for (int off = 16; off; off >>= 1) mx = fmaxf(mx, __shfl_xor(mx, off, 32));
      float sum = 0.f;
      for (int c = lane; c < S_LEN; c += 32) {
        float e = __expf(row[c] - mx);
        row[c] = e;
        sum += e;
      }
      for (int off = 16; off; off >>= 1) sum += __shfl_xor(sum, off, 32);
      float inv = 1.f / sum;
      float* mrow = mn + rr * SPITCH;
      for (int c = lane; c < S_LEN; c += 32) {
        float wv = row[c] * inv;
        row[c] = wv;
        mrow[c] += wv * 0.125f;           // 1/NH
      }
    }
    __syncthreads();

    // ---- Phase C: O(16x64) = P(16x2048) @ V(2048x64), K split over waves -
    {
      int tn = wave & 3;                   // which 16-col slab of head dim
      int kh = wave >> 2;                  // which half of K=2048
      int n0h = tn * 16;
      v8f acc = {0.f,0.f,0.f,0.f,0.f,0.f,0.f,0.f};
      int kbeg = kh * 1024, kend = kbeg + 1024;
      for (int k = kbeg; k < kend; k += 4) {
        v2f a = *(const v2f*)(sc + l * SPITCH + k + half * 2);
        const float* vr = proj + (size_t)(b * S_LEN + k + half * 2) * TH3
                               + 2 * HID + h * HD + n0h + l;
        v2f bf = { vr[0], vr[TH3] };       // V rows k+half*2, k+half*2+1
        acc = wmma_f32(a, bf, acc);
      }
      if (kh == 1) {
#pragma unroll
        for (int r = 0; r < 8; ++r) red[tn * 256 + r * 32 + lane] = acc[r];
      }
      __syncthreads();
      if (kh == 0) {
#pragma unroll
        for (int r = 0; r < 8; ++r) {
          float v = acc[r] + red[tn * 256 + r * 32 + lane];
          int m = r + half * 8;
          ctx[(size_t)(b * S_LEN + q0 + m) * HID + h * HD + n0h + l] = v;
        }
      }
      __syncthreads();                     // sc reused by next head's Phase A
    }
  }

  // ---- emit mean attention weights [B,S,S] -------------------------------
  for (int i = tid; i < 16 * S_LEN; i += 256) {
    int m = i >> 11, c = i & (S_LEN - 1);
    out2[(size_t)(b * S_LEN + q0 + m) * S_LEN + c] = mn[m * SPITCH + c];
  }
}

// ---------------------------------------------------------------------------
extern "C" void kernel_launch(void* const* d_in, const int* in_sizes, int n_in,
                              void* d_out, int out_size, void* d_ws, size_t ws_size,
                              hipStream_t stream) {
  (void)in_sizes; (void)n_in; (void)out_size; (void)ws_size;
  const float* x     = (const float*)d_in[0];   // [4,2048,512]
  const float* posi  = (const float*)d_in[1];   // [4,2048,3]
  const int*   mask  = (const int*)  d_in[2];   // [4,2048]
  const float* w_qkv = (const float*)d_in[3];   // [1536,512]
  const float* b_qkv = (const float*)d_in[4];   // [1536]
  const float* w_out = (const float*)d_in[5];   // [512,512]
  const float* b_out = (const float*)d_in[6];   // [512]
  float* out = (float*)d_out;

  const int M = 4 * S_LEN;                      // 8192
  float* proj = (float*)d_ws;                   // [8192,1536] = 48 MB
  float* ctx  = proj + (size_t)M * TH3;         // [8192,512]  = 16 MB

  // 1) QKV projection: proj = x @ w_qkv^T + b_qkv   (32x32 tiles per wave)
  {
    int tiles = (M / 32) * (TH3 / 32);          // 12288 wave-tiles
    gemm32_bias<<<tiles / 8, 256, 0, stream>>>(x, w_qkv, b_qkv, proj, M, TH3, HID);
  }

  // 2) Fused attention (scores+bias+mask+softmax+mean+PV) in 261 KB LDS
  {
    size_t shmem = (size_t)(16 * SPITCH * 2 + 1024 + 48) * sizeof(float);
    attn_fused<<<dim3(S_LEN / 16, 4), 256, shmem, stream>>>(
        proj, posi, mask, ctx, out + (size_t)M * HID);
  }

  // 3) Output projection: out = ctx @ w_out^T + b_out
  {
    int tiles = (M / 32) * (HID / 32);          // 4096 wave-tiles
    gemm32_bias<<<tiles / 8, 256, 0, stream>>>(ctx, w_out, b_out, out, M, HID, HID);
  }
}